// RiemannianTokenizer_24129126269379
// MI455X (gfx1250) — compile-verified
//
#include <hip/hip_runtime.h>
#include <hip/hip_bf16.h>
#include <math.h>

// ---- problem constants (from reference) ----
#define C64     64
#define WINDOW  100
#define STRIDE  50
#define EMBED   32
#define NTOK    64
#define TDIM    2080        // 64*65/2
#define LDM     65          // padded LDS stride for 64x64 matrices
#define LDX     104         // padded LDS stride for the 64x100 window
#define NSQRT   4           // inverse-scaling-and-squaring depth
#define NS_IT   8           // Newton-Schulz iterations per sqrt stage
#define SER_M   7           // log(I+X) series terms
#define NS_IT_A 30          // Newton-Schulz iterations for G^{-1/2}

typedef float v2f __attribute__((ext_vector_type(2)));
typedef float v8f __attribute__((ext_vector_type(8)));

__device__ __forceinline__ v8f wmma4(v2f a, v2f b, v8f c) {
  // D = A(16x4,f32) * B(4x16,f32) + C(16x16,f32)
  return __builtin_amdgcn_wmma_f32_16x16x4_f32(
      /*neg_a=*/false, a, /*neg_b=*/false, b,
      /*c_mod=*/(short)0, c, /*reuse_a=*/false, /*reuse_b=*/false);
}

// C(64x64,ldc) = alpha * A(64xK, lda) * B(Kx64, ldb)  [+ diag_add on the diagonal]
// 256 threads (8 waves); each wave computes two 16x16 tiles. K % 4 == 0.
__device__ __forceinline__ void gemm64(const float* __restrict__ A, int lda,
                                       const float* __restrict__ B, int ldb,
                                       float* __restrict__ Cm, int ldc,
                                       int K, float alpha, float diag_add)
{
  const int tid  = threadIdx.x;
  const int lane = tid & 31;
  const int wid  = tid >> 5;
  const int hlf  = lane >> 4;   // 0: K=k0,k0+1 ; 1: K=k0+2,k0+3
  const int l16  = lane & 15;
  for (int t = wid; t < 16; t += 8) {
    const int tm = t >> 2, tn = t & 3;
    v8f acc = {};
    const float* Ar = A + (tm * 16 + l16) * lda + 2 * hlf;     // row of A
    const float* Bc = B + (2 * hlf) * ldb + tn * 16 + l16;     // column of B
    for (int k0 = 0; k0 < K; k0 += 4) {
      v2f a, b;
      a.x = Ar[k0];            a.y = Ar[k0 + 1];
      b.x = Bc[k0 * ldb];      b.y = Bc[(k0 + 1) * ldb];
      acc = wmma4(a, b, acc);
    }
    #pragma unroll
    for (int r = 0; r < 8; ++r) {
      const int row = tm * 16 + 8 * hlf + r;
      const int col = tn * 16 + l16;
      float v = acc[r] * alpha;
      if (row == col) v += diag_add;
      Cm[row * ldc + col] = v;
    }
  }
}

// G(64x64,ldg) = X(64xK,ldx) * X^T  via WMMA (K % 4 == 0)
__device__ __forceinline__ void gram64(const float* __restrict__ X, int ldx,
                                       float* __restrict__ G, int ldg, int K)
{
  const int tid  = threadIdx.x;
  const int lane = tid & 31;
  const int wid  = tid >> 5;
  const int hlf  = lane >> 4;
  const int l16  = lane & 15;
  for (int t = wid; t < 16; t += 8) {
    const int tm = t >> 2, tn = t & 3;
    v8f acc = {};
    const float* Ar = X + (tm * 16 + l16) * ldx + 2 * hlf;
    const float* Br = X + (tn * 16 + l16) * ldx + 2 * hlf;  // B[k][n] = X[n][k]
    for (int k0 = 0; k0 < K; k0 += 4) {
      v2f a, b;
      a.x = Ar[k0]; a.y = Ar[k0 + 1];
      b.x = Br[k0]; b.y = Br[k0 + 1];
      acc = wmma4(a, b, acc);
    }
    #pragma unroll
    for (int r = 0; r < 8; ++r) {
      const int row = tm * 16 + 8 * hlf + r;
      const int col = tn * 16 + l16;
      G[row * ldg + col] = acc[r];
    }
  }
}

__device__ __forceinline__ float block_reduce_sum(float v, float* red) {
  const int tid = threadIdx.x;
  __syncthreads();
  for (int off = 16; off; off >>= 1) v += __shfl_xor(v, off, 32);
  if ((tid & 31) == 0) red[tid >> 5] = v;
  __syncthreads();
  if (tid == 0) {
    float s = 0.f;
    #pragma unroll
    for (int i = 0; i < 8; ++i) s += red[i];
    red[0] = s;
  }
  __syncthreads();
  float s = red[0];
  __syncthreads();
  return s;
}

// ---- one-time kernel: W = reference_mean^{-1/2} via Newton-Schulz ----
__global__ void __launch_bounds__(256)
prep_whitener(const float* __restrict__ G, float* __restrict__ W)
{
  extern __shared__ float smem[];
  float* Y   = smem;
  float* Z   = Y + 64 * LDM;
  float* M   = Z + 64 * LDM;
  float* Tb  = M + 64 * LDM;
  float* red = Tb + 64 * LDM;
  const int tid = threadIdx.x;

  float part = 0.f;
  for (int i = tid; i < 4096; i += 256) { float g = G[i]; part += g * g; }
  const float c = sqrtf(block_reduce_sum(part, red));   // >= lambda_max
  const float invc = 1.0f / c;

  for (int i = tid; i < 4096; i += 256) {
    const int r = i >> 6, cc = i & 63;
    Y[r * LDM + cc] = G[i] * invc;                 // A = G/c, eigs in (0,1]
    Z[r * LDM + cc] = (r == cc) ? 1.0f : 0.0f;
  }
  __syncthreads();
  for (int it = 0; it < NS_IT_A; ++it) {
    gemm64(Z, LDM, Y, LDM, M,  LDM, 64, -0.5f, 1.5f); __syncthreads(); // M=(3I-ZY)/2
    gemm64(Y, LDM, M, LDM, Tb, LDM, 64,  1.0f, 0.0f); __syncthreads(); // Y' = Y M
    gemm64(M, LDM, Z, LDM, Y,  LDM, 64,  1.0f, 0.0f); __syncthreads(); // Z' = M Z
    float* oy = Y; Y = Tb; Tb = Z; Z = oy;
  }
  const float s = rsqrtf(c);                       // W = Z / sqrt(c) -> G^{-1/2}
  for (int i = tid; i < 4096; i += 256)
    W[i] = Z[(i >> 6) * LDM + (i & 63)] * s;
}

// ---- per-window kernel: cov -> whiten -> logm (NS) -> tangent -> proj -> VQ ----
__global__ void __launch_bounds__(256)
tokenize_windows(const float* __restrict__ data, long long T, int NW,
                 const float* __restrict__ Wg, const float* __restrict__ Pw,
                 const float* __restrict__ Pb, const float* __restrict__ CB,
                 float* __restrict__ out)
{
  extern __shared__ float smem[];
  float* Xb  = smem;                 // 64 x LDX
  float* Wl  = Xb + 64 * LDX;        // 64 x LDM
  float* B0  = Wl + 64 * LDM;
  float* B1  = B0 + 64 * LDM;
  float* B2  = B1 + 64 * LDM;
  float* B3  = B2 + 64 * LDM;
  float* mu  = B3 + 64 * LDM;        // 64
  float* red = mu + 64;              // 64
  float* zs  = red + 64;             // 32
  float* d2  = zs + 32;              // 64
  float* sc  = d2 + 64;              // 4 (padding)
  unsigned short* tbl = (unsigned short*)(sc + 4);   // 2080 packed (i<<8|j)

  const int tid  = threadIdx.x;
  const int lane = tid & 31;
  const int wid  = tid >> 5;
  const int w    = blockIdx.x;
  const long long base = (long long)w * STRIDE;

  // prefetch projection matrix (resident in L2 across the grid)
  __builtin_prefetch(Pw + (size_t)tid * 260, 0, 1);

  // stage the 64x100 window and the whitener into LDS
  for (int i = tid; i < 64 * WINDOW; i += 256) {
    const int r = i / WINDOW, k = i % WINDOW;
    Xb[r * LDX + k] = data[(long long)r * T + base + k];
  }
  for (int i = tid; i < 4096; i += 256)
    Wl[(i >> 6) * LDM + (i & 63)] = Wg[i];
  if (tid < 64) {                    // upper-tri index table
    const int i0 = tid;
    int q = i0 * 64 - (i0 * (i0 - 1)) / 2;
    for (int j = i0; j < 64; ++j) tbl[q++] = (unsigned short)((i0 << 8) | j);
  }
  __syncthreads();

  // window mean
  if (tid < 64) {
    float s = 0.f;
    for (int k = 0; k < WINDOW; ++k) s += Xb[tid * LDX + k];
    mu[tid] = s * (1.0f / WINDOW);
  }
  __syncthreads();

  // Sigma = (X X^T - N mu mu^T) / (N-1)
  gram64(Xb, LDX, B0, LDM, WINDOW);
  __syncthreads();
  for (int i = tid; i < 4096; i += 256) {
    const int r = i >> 6, cc = i & 63;
    B0[r * LDM + cc] = (B0[r * LDM + cc] - (float)WINDOW * mu[r] * mu[cc])
                       * (1.0f / (WINDOW - 1));
  }
  __syncthreads();

  // S = W Sigma W
  gemm64(Wl, LDM, B0, LDM, B1, LDM, 64, 1.0f, 0.0f); __syncthreads();
  gemm64(B1, LDM, Wl, LDM, B0, LDM, 64, 1.0f, 0.0f); __syncthreads();

  // normalize: A = S / ||S||_F  (eigs in (0,1])
  float part = 0.f;
  for (int i = tid; i < 4096; i += 256) {
    const float v = B0[(i >> 6) * LDM + (i & 63)];
    part += v * v;
  }
  const float c = sqrtf(block_reduce_sum(part, red));
  const float invc = 1.0f / c;
  for (int i = tid; i < 4096; i += 256) B0[(i >> 6) * LDM + (i & 63)] *= invc;
  __syncthreads();

  // inverse scaling & squaring: NSQRT Newton-Schulz matrix square roots
  float* Y = B0; float* Z = B1; float* M = B2; float* Tb = B3;
  for (int st = 0; st < NSQRT; ++st) {
    for (int i = tid; i < 4096; i += 256) {
      const int r = i >> 6, cc = i & 63;
      Z[r * LDM + cc] = (r == cc) ? 1.0f : 0.0f;
    }
    __syncthreads();
    for (int it = 0; it < NS_IT; ++it) {
      gemm64(Z, LDM, Y, LDM, M,  LDM, 64, -0.5f, 1.5f); __syncthreads();
      gemm64(Y, LDM, M, LDM, Tb, LDM, 64,  1.0f, 0.0f); __syncthreads();
      gemm64(M, LDM, Z, LDM, Y,  LDM, 64,  1.0f, 0.0f); __syncthreads();
      float* oy = Y; Y = Tb; Tb = Z; Z = oy;
    }
    // Y now holds A^{1/2}; next stage reuses it as A
  }

  // X = A^{1/2^NSQRT} - I (in place in Y)
  if (tid < 64) Y[tid * LDM + tid] -= 1.0f;
  __syncthreads();

  // log(I+X) via Horner: B = c_M I; B = X*B + c_m I; ... ; L = X*B
  float* Bacc = Z; float* Tmp = Tb;
  {
    const float cM = ((SER_M & 1) ? 1.0f : -1.0f) / (float)SER_M;
    for (int i = tid; i < 4096; i += 256) {
      const int r = i >> 6, cc = i & 63;
      Bacc[r * LDM + cc] = (r == cc) ? cM : 0.0f;
    }
  }
  __syncthreads();
  for (int m = SER_M - 1; m >= 1; --m) {
    const float cm = ((m & 1) ? 1.0f : -1.0f) / (float)m;
    gemm64(Y, LDM, Bacc, LDM, Tmp, LDM, 64, 1.0f, cm); __syncthreads();
    float* o = Bacc; Bacc = Tmp; Tmp = o;
  }
  // logS = 2^NSQRT * (X * Bacc) + log(c) I
  gemm64(Y, LDM, Bacc, LDM, M, LDM, 64, (float)(1 << NSQRT), logf(c));
  __syncthreads();
  const float* Lg = M;

  // tangent vector (upper tri, sqrt(2) off-diag) -> z = Pw @ t + Pb
  float acc[4] = {0.f, 0.f, 0.f, 0.f};
  const float SQ2 = 1.41421356237309515f;
  for (int q = lane; q < TDIM; q += 32) {
    const unsigned short ij = tbl[q];
    const int i0 = ij >> 8, j0 = ij & 255;
    const float v = Lg[i0 * LDM + j0] * ((i0 == j0) ? 1.0f : SQ2);
    #pragma unroll
    for (int e = 0; e < 4; ++e)
      acc[e] += v * Pw[(size_t)(wid * 4 + e) * TDIM + q];
  }
  #pragma unroll
  for (int e = 0; e < 4; ++e) {
    float a = acc[e];
    for (int off = 16; off; off >>= 1) a += __shfl_xor(a, off, 32);
    if (lane == 0) {
      const float zv = a + Pb[wid * 4 + e];
      zs[wid * 4 + e] = zv;
      out[(long long)NW + (long long)w * EMBED + (wid * 4 + e)] = zv;
    }
  }
  __syncthreads();

  // VQ: nearest codebook entry (first-min tie-break == reference argmax(-d2))
  if (tid < NTOK) {
    float d = 0.f;
    for (int k = 0; k < EMBED; ++k) {
      const float df = zs[k] - CB[tid * EMBED + k];
      d += df * df;
    }
    d2[tid] = d;
  }
  __syncthreads();
  if (tid == 0) {
    int best = 0; float bd = d2[0];
    for (int k = 1; k < NTOK; ++k) if (d2[k] < bd) { bd = d2[k]; best = k; }
    out[w] = (float)best;
  }
}

extern "C" void kernel_launch(void* const* d_in, const int* in_sizes, int n_in,
                              void* d_out, int out_size, void* d_ws, size_t ws_size,
                              hipStream_t stream) {
  const float* data = (const float*)d_in[0];   // (64, T)
  const float* G    = (const float*)d_in[1];   // (64, 64)
  const float* Pw   = (const float*)d_in[2];   // (32, 2080)
  const float* Pb   = (const float*)d_in[3];   // (32,)
  const float* CB   = (const float*)d_in[4];   // (64, 32)

  const long long T = (long long)in_sizes[0] / C64;
  const int NW = (int)((T - WINDOW) / STRIDE + 1);

  float* Wws = (float*)d_ws;                   // 4096 floats: G^{-1/2}

  const size_t smemA = (size_t)(4 * 64 * LDM + 64) * sizeof(float);
  prep_whitener<<<1, 256, smemA, stream>>>(G, Wws);

  const size_t smemB = (size_t)(64 * LDX + 5 * 64 * LDM + 64 + 64 + 32 + 64 + 4)
                         * sizeof(float)
                     + (size_t)TDIM * sizeof(unsigned short);
  tokenize_windows<<<NW, 256, smemB, stream>>>(data, T, NW, Wws, Pw, Pb, CB,
                                               (float*)d_out);
}